// GCN_2834678415392
// MI455X (gfx1250) — compile-verified
//
#include <hip/hip_runtime.h>
#include <math.h>

typedef __attribute__((ext_vector_type(2))) float v2f;
typedef __attribute__((ext_vector_type(8))) float v8f;

// ---------------- degree / normalization ----------------

__global__ void k_init_deg(float* __restrict__ deg, int n) {
    int i = blockIdx.x * blockDim.x + threadIdx.x;
    if (i < n) deg[i] = 1.0f;  // self-loop contributes 1 to every node's degree
}

__global__ void k_accum_deg(const int* __restrict__ col, float* __restrict__ deg, int e) {
    int i = blockIdx.x * blockDim.x + threadIdx.x;
    if (i < e) atomicAdd(&deg[col[i]], 1.0f);
}

__global__ void k_rsqrt_inplace(float* __restrict__ deg, int n) {
    int i = blockIdx.x * blockDim.x + threadIdx.x;
    if (i < n) deg[i] = rsqrtf(deg[i]);  // deg >= 1 always (self-loops)
}

// ---------------- fp32 WMMA GEMM: C[M x NC] = A[M x K] @ B[K x NB], NB,NC <= 16 ------------
// One wave32 computes a 16x16 tile with V_WMMA_F32_16X16X4_F32.
// B is staged once per block into LDS, transposed + zero-padded to 16 columns, with a
// 2-DWORD row pad to avoid 16-way bank conflicts. All loads are unconditional:
// out-of-range A rows are clamped to M-1 (their D rows are never stored, and A-tile
// row t only feeds D-tile row t, so no corruption of stored rows).

template <int K, int NB, int NC>
__global__ void k_gemm16_wmma(const float* __restrict__ A, const float* __restrict__ B,
                              float* __restrict__ C, int M) {
    constexpr int KP = K + 2;            // padded LDS row stride (keeps 8B alignment, breaks bank conflict)
    __shared__ float Bt[16 * KP];        // Bt[n*KP + k] = (n < NB) ? B[k*NB + n] : 0
    const int tid = threadIdx.x;

    for (int idx = tid; idx < 16 * K; idx += blockDim.x) {
        int n = idx / K;
        int k = idx - n * K;
        Bt[n * KP + k] = (n < NB) ? B[k * NB + n] : 0.0f;
    }
    __syncthreads();

    const int lane = tid & 31;
    const int m0 = (blockIdx.x * (blockDim.x >> 5) + (tid >> 5)) << 4;
    if (m0 >= M) return;                 // wave-uniform exit: EXEC all-ones for WMMA

    const int half = lane >> 4;          // 0: lanes 0-15, 1: lanes 16-31
    const int lidx = lane & 15;

    int mA = m0 + lidx;
    if (mA >= M) mA = M - 1;             // clamp instead of predicating the loads
    const float* __restrict__ arow = A + (unsigned)mA * (unsigned)K;
    const float* __restrict__ brow = Bt + lidx * KP;

    v8f acc = {};
#pragma unroll
    for (int k0 = 0; k0 < K; k0 += 4) {
        const int ka = k0 + (half << 1);
        // A 16x4: lane l, VGPR v -> A[l%16][k0 + v + 2*(l/16)]  (a.x,a.y contiguous -> b64 load)
        v2f a = *(const v2f*)(arow + ka);
        // B 4x16: lane l, VGPR v -> B[k0 + v + 2*(l/16)][l%16]  (contiguous in transposed LDS)
        v2f b = *(const v2f*)(brow + ka);
        acc = __builtin_amdgcn_wmma_f32_16x16x4_f32(
            /*neg_a=*/false, a, /*neg_b=*/false, b,
            /*c_mod=*/(short)0, acc, /*reuse_a=*/false, /*reuse_b=*/false);
    }

    // D 16x16: lane l, VGPR v -> row (v + 8*(l/16)), col (l%16)
    if (lidx < NC) {
#pragma unroll
        for (int v = 0; v < 8; ++v) {
            int m = m0 + v + (half << 3);
            if (m < M) C[(unsigned)m * (unsigned)NC + lidx] = acc[v];
        }
    }
}

// ---------------- self-loop init: out[i,f] = h[i,f] * dinv[i]^2 ----------------

__global__ void k_selfloop(const float* __restrict__ h, const float* __restrict__ dinv,
                           float* __restrict__ out, int n, int F) {
    long long i = blockIdx.x * (long long)blockDim.x + threadIdx.x;
    long long t = (long long)n * F;
    if (i >= t) return;
    int node = (int)(i / F);
    float d = dinv[node];
    out[i] = h[i] * d * d;
}

// ---------------- edge scatter, 16-wide features: 4 lanes/edge, float4 gathers ----------------

__global__ void k_scatter16(const int* __restrict__ row, const int* __restrict__ col,
                            const float* __restrict__ dinv, const float* __restrict__ h,
                            float* __restrict__ out, int e) {
    long long gid = blockIdx.x * (long long)blockDim.x + threadIdx.x;
    if (gid >= (long long)e * 4) return;
    int ed  = (int)(gid >> 2);
    int sub = (int)(gid & 3);
    int r = row[ed], c = col[ed];
    float w = dinv[r] * dinv[c];
    float4 v = *(const float4*)(h + (size_t)r * 16 + sub * 4);
    float* op = out + (size_t)c * 16 + sub * 4;
    atomicAdd(op + 0, v.x * w);
    atomicAdd(op + 1, v.y * w);
    atomicAdd(op + 2, v.z * w);
    atomicAdd(op + 3, v.w * w);
}

// ---------------- edge scatter, 10-wide features: 1 lane per (edge, class) ----------------

__global__ void k_scatter10(const int* __restrict__ row, const int* __restrict__ col,
                            const float* __restrict__ dinv, const float* __restrict__ h,
                            float* __restrict__ out, int e) {
    long long gid = blockIdx.x * (long long)blockDim.x + threadIdx.x;
    if (gid >= (long long)e * 10) return;
    int ed = (int)(gid / 10);
    int j  = (int)(gid - (long long)ed * 10);
    int r = row[ed], c = col[ed];
    float w = dinv[r] * dinv[c];
    atomicAdd(&out[(size_t)c * 10 + j], h[(size_t)r * 10 + j] * w);
}

// ---------------- bias + ReLU (in place, 16-wide) ----------------

__global__ void k_relu_bias16(float* __restrict__ x, const float* __restrict__ b, int n) {
    long long i = blockIdx.x * (long long)blockDim.x + threadIdx.x;
    if (i >= (long long)n * 16) return;
    int f = (int)(i & 15);
    float v = x[i] + b[f];
    x[i] = v > 0.0f ? v : 0.0f;
}

// ---------------- bias + log_softmax over 10 classes ----------------

__global__ void k_logsoftmax10(const float* __restrict__ z, const float* __restrict__ b,
                               float* __restrict__ y, int n) {
    int i = blockIdx.x * blockDim.x + threadIdx.x;
    if (i >= n) return;
    float l[10];
    float m = -3.402823466e+38f;
#pragma unroll
    for (int j = 0; j < 10; ++j) {
        l[j] = z[(size_t)i * 10 + j] + b[j];
        m = fmaxf(m, l[j]);
    }
    float s = 0.0f;
#pragma unroll
    for (int j = 0; j < 10; ++j) s += expf(l[j] - m);
    float ls = logf(s) + m;
#pragma unroll
    for (int j = 0; j < 10; ++j) y[(size_t)i * 10 + j] = l[j] - ls;
}

// ---------------- launch ----------------

extern "C" void kernel_launch(void* const* d_in, const int* in_sizes, int n_in,
                              void* d_out, int out_size, void* d_ws, size_t ws_size,
                              hipStream_t stream) {
    const float* x  = (const float*)d_in[0];
    const int*   ei = (const int*)d_in[1];
    const float* W1 = (const float*)d_in[2];
    const float* b1 = (const float*)d_in[3];
    const float* W2 = (const float*)d_in[4];
    const float* b2 = (const float*)d_in[5];

    const int NFEAT = 128, HID = 16, NCLS = 10;
    int N = in_sizes[0] / NFEAT;   // 100000
    int E = in_sizes[1] / 2;       // 3200000
    const int* row = ei;           // edge_index[0] : source nodes
    const int* col = ei + E;       // edge_index[1] : target nodes

    // workspace layout (floats): dinv[N] | h[N*16] | out1[N*16] | h2[N*10] | out2[N*10]
    float* ws   = (float*)d_ws;
    float* dinv = ws;
    float* h    = dinv + N;
    float* out1 = h + (size_t)N * HID;
    float* h2   = out1 + (size_t)N * HID;
    float* out2 = h2 + (size_t)N * NCLS;

    const int B = 256;
    auto cdiv = [](long long a, long long b) { return (int)((a + b - 1) / b); };

    // normalization: deg (with self-loop) -> dinv = rsqrt(deg)
    k_init_deg<<<cdiv(N, B), B, 0, stream>>>(dinv, N);
    k_accum_deg<<<cdiv(E, B), B, 0, stream>>>(col, dinv, E);
    k_rsqrt_inplace<<<cdiv(N, B), B, 0, stream>>>(dinv, N);

    int waves = cdiv(N, 16);
    int gemmBlocks = cdiv((long long)waves * 32, B);

    // layer 1: h = x @ W1 ; out1 = D^-1/2 (A+I) D^-1/2 h ; relu(out1 + b1) in place
    k_gemm16_wmma<128, 16, 16><<<gemmBlocks, B, 0, stream>>>(x, W1, h, N);
    k_selfloop<<<cdiv((long long)N * HID, B), B, 0, stream>>>(h, dinv, out1, N, HID);
    k_scatter16<<<cdiv((long long)E * 4, B), B, 0, stream>>>(row, col, dinv, h, out1, E);
    k_relu_bias16<<<cdiv((long long)N * HID, B), B, 0, stream>>>(out1, b1, N);

    // layer 2: h2 = h1 @ W2 ; out2 = propagate ; log_softmax(out2 + b2)
    k_gemm16_wmma<16, 10, 10><<<gemmBlocks, B, 0, stream>>>(out1, W2, h2, N);
    k_selfloop<<<cdiv((long long)N * NCLS, B), B, 0, stream>>>(h2, dinv, out2, N, NCLS);
    k_scatter10<<<cdiv((long long)E * NCLS, B), B, 0, stream>>>(row, col, dinv, h2, out2, E);
    k_logsoftmax10<<<cdiv(N, B), B, 0, stream>>>(out2, b2, (float*)d_out, N);
}